// RotaryEmbedding_62191126446756
// MI455X (gfx1250) — compile-verified
//
#include <hip/hip_runtime.h>
#include <math.h>

// ---------------------------------------------------------------------------
// RoPE for qkv (4, 4096, 3, 16, 128) fp32, seqlen offsets (4,) int32.
// Memory-bound: ~0.8 GB moved -> ~35us floor at 23.3 TB/s. Strategy:
//   1) tiny table kernel: precise sincosf for (4096+512) x 64 positions/freqs
//      into d_ws (2.36 MB, L2-resident on MI455X's 192 MB L2).
//   2) streaming kernel: one wave32 per 128-float row, b128 loads/stores with
//      NT temporal hints (stream touched once -> don't evict the hot table),
//      wave32 shfl_xor(16) to pair x[f] with x[f+64].
// NOTE: rope_apply_table is defined FIRST so the disasm snippet shows the hot
// streaming kernel (verify th:NT b128 ops + shuffle lowering).
// ---------------------------------------------------------------------------

// native clang vector: required by __builtin_nontemporal_load/store,
// 16-byte aligned, lowers to global_load_b128 / global_store_b128
typedef float v4f __attribute__((ext_vector_type(4)));

#define B_    4
#define S_    4096
#define T_    3
#define H_    16
#define D_    128
#define HALF_ 64
#define MAX_OFFSET_ 512
#define MAXPOS_ (S_ + MAX_OFFSET_)          // 4608
#define NTAB_   (MAXPOS_ * HALF_)           // 294912 entries per table
#define ROWS_   (B_ * S_ * T_ * H_)         // 786432 rows of 128 floats

// ---------------- main streaming kernel (table path) -----------------
__global__ __launch_bounds__(256) void rope_apply_table(const float* __restrict__ qkv,
                                                        const int*   __restrict__ offs,
                                                        const float* __restrict__ cosT,
                                                        const float* __restrict__ sinT,
                                                        float*       __restrict__ out) {
    const int lane = threadIdx.x & 31;
    const long long row = (long long)blockIdx.x * 8 + (threadIdx.x >> 5);
    if (row >= ROWS_) return;

    // row -> (b, s, t, h); memory order matches row order (h fastest)
    long long r1 = row / H_;
    const int t = (int)(r1 % T_);
    long long r2 = r1 / T_;
    const int s = (int)(r2 % S_);
    const int b = (int)(r2 / S_);

    const v4f* __restrict__ src = (const v4f*)(qkv + row * D_);
    v4f*       __restrict__ dst = (v4f*)(out + row * D_);

    // stream data: read once, NT hint keeps it from claiming L2 residency
    v4f x = __builtin_nontemporal_load(src + lane);      // global_load_b128 th:NT

    if (t == 2) {                         // v: straight copy (whole wave uniform)
        __builtin_nontemporal_store(x, dst + lane);
        return;
    }

    // partner holds x[f +/- 64] (lanes 0-15 = first half, 16-31 = second half)
    v4f px;
    px.x = __shfl_xor(x.x, 16);
    px.y = __shfl_xor(x.y, 16);
    px.z = __shfl_xor(x.z, 16);
    px.w = __shfl_xor(x.w, 16);

    // wave-uniform batch offset -> scalar broadcast
    const int pos   = __builtin_amdgcn_readfirstlane(offs[b]) + s;
    const int fbase = (lane & 15) * 4;    // 4 consecutive frequency indices
    // table loads: default (RT) policy -> stays hot in L2/WGP$, reused heavily
    const v4f c4 = *(const v4f*)(cosT + pos * HALF_ + fbase);
    const v4f s4 = *(const v4f*)(sinT + pos * HALF_ + fbase);

    // first half: x*cos - partner*sin ; second half: x*cos + partner*sin
    const float sgn = (lane < 16) ? -1.0f : 1.0f;

    v4f r;
    r.x = fmaf(x.x, c4.x, sgn * (px.x * s4.x));
    r.y = fmaf(x.y, c4.y, sgn * (px.y * s4.y));
    r.z = fmaf(x.z, c4.z, sgn * (px.z * s4.z));
    r.w = fmaf(x.w, c4.w, sgn * (px.w * s4.w));

    __builtin_nontemporal_store(r, dst + lane);          // global_store_b128 th:NT
}

// ---------------- table builder: precise sin/cos -----------------
__global__ __launch_bounds__(256) void rope_table_kernel(float* __restrict__ cosT,
                                                         float* __restrict__ sinT) {
    int tid = blockIdx.x * blockDim.x + threadIdx.x;
    if (tid >= NTAB_) return;
    int p = tid >> 6;          // position
    int f = tid & 63;          // frequency index
    // inv_freq = 10000^(-f/64), matching the fp32 reference tables
    float inv_freq = powf(10000.0f, -(float)f * (1.0f / 64.0f));
    float angle = (float)p * inv_freq;
    float s, c;
    sincosf(angle, &s, &c);    // precise (Payne-Hanek) - only ~295K calls
    cosT[tid] = c;
    sinT[tid] = s;
}

// ---------------- fallback: inline trig (only if ws too small) -----------------
__global__ __launch_bounds__(256) void rope_apply_inline(const float* __restrict__ qkv,
                                                         const int*   __restrict__ offs,
                                                         float*       __restrict__ out) {
    const int lane = threadIdx.x & 31;
    const long long row = (long long)blockIdx.x * 8 + (threadIdx.x >> 5);
    if (row >= ROWS_) return;

    long long r1 = row / H_;
    const int t = (int)(r1 % T_);
    long long r2 = r1 / T_;
    const int s = (int)(r2 % S_);
    const int b = (int)(r2 / S_);

    const v4f* __restrict__ src = (const v4f*)(qkv + row * D_);
    v4f*       __restrict__ dst = (v4f*)(out + row * D_);

    v4f x = __builtin_nontemporal_load(src + lane);
    if (t == 2) { __builtin_nontemporal_store(x, dst + lane); return; }

    v4f px;
    px.x = __shfl_xor(x.x, 16);
    px.y = __shfl_xor(x.y, 16);
    px.z = __shfl_xor(x.z, 16);
    px.w = __shfl_xor(x.w, 16);

    const int   pos   = __builtin_amdgcn_readfirstlane(offs[b]) + s;
    const int   fbase = (lane & 15) * 4;
    const float sgn   = (lane < 16) ? -1.0f : 1.0f;

    float xs[4] = {x.x, x.y, x.z, x.w};
    float ps[4] = {px.x, px.y, px.z, px.w};
    float rs[4];
#pragma unroll
    for (int c = 0; c < 4; ++c) {
        float inv_freq = powf(10000.0f, -(float)(fbase + c) * (1.0f / 64.0f));
        float sv, cv;
        sincosf((float)pos * inv_freq, &sv, &cv);
        rs[c] = fmaf(xs[c], cv, sgn * (ps[c] * sv));
    }
    v4f r;
    r.x = rs[0]; r.y = rs[1]; r.z = rs[2]; r.w = rs[3];
    __builtin_nontemporal_store(r, dst + lane);
}

extern "C" void kernel_launch(void* const* d_in, const int* in_sizes, int n_in,
                              void* d_out, int out_size, void* d_ws, size_t ws_size,
                              hipStream_t stream) {
    const float* qkv  = (const float*)d_in[0];
    const int*   offs = (const int*)d_in[1];
    float*       out  = (float*)d_out;

    const size_t tabBytes = (size_t)2 * NTAB_ * sizeof(float);   // 2.36 MB

    const int rowBlocks = ROWS_ / 8;                             // 8 waves/block

    if (d_ws != nullptr && ws_size >= tabBytes) {
        float* cosT = (float*)d_ws;
        float* sinT = cosT + NTAB_;
        rope_table_kernel<<<(NTAB_ + 255) / 256, 256, 0, stream>>>(cosT, sinT);
        rope_apply_table<<<rowBlocks, 256, 0, stream>>>(qkv, offs, cosT, sinT, out);
    } else {
        rope_apply_inline<<<rowBlocks, 256, 0, stream>>>(qkv, offs, out);
    }
}